// CXBPU_55611236549128
// MI455X (gfx1250) — compile-verified
//
#include <hip/hip_runtime.h>

typedef __attribute__((ext_vector_type(2))) float v2f;
typedef __attribute__((ext_vector_type(4))) float v4f;
typedef __attribute__((ext_vector_type(8))) float v8f;

#define N_HID   2048
#define BATCH   64
#define TSTEPS  128
#define IN_DIM  4
#define OUT_DIM 2
#define NSENS   256
#define NOUT    128
#define KMICRO  4

#define COLS_PER_BLK 32
#define NBLK   (N_HID / COLS_PER_BLK)   // 64 blocks, one per WGP
#define THREADS 256                     // 8 wave32
#define LDS_STRIDE 2052                 // 2048 + 4-word pad -> conflict-free b64 reads

#define KBODY   32                      // K-values per pipelined body (8 WMMA quads)
#define NBODY   (N_HID / KBODY)         // 64 bodies

// ---------------- init kernels ----------------
__global__ void CXBPU_init_kernel(float* __restrict__ h0, unsigned* __restrict__ cnt,
                                  int* __restrict__ inv, int* __restrict__ oidx32,
                                  const long long* __restrict__ oidx64) {
    int i = blockIdx.x * blockDim.x + threadIdx.x;
    if (i < BATCH * N_HID) h0[i] = 0.0f;
    if (i < N_HID)         inv[i] = -1;
    if (i < NOUT)          oidx32[i] = (int)oidx64[i];
    if (i == 0)            *cnt = 0u;
}

__global__ void CXBPU_scatter_kernel(const long long* __restrict__ sens,
                                     int* __restrict__ inv) {
    int s = threadIdx.x;                 // 256 threads
    if (s < NSENS) inv[(int)sens[s]] = s;
}

// 8 chained WMMA quads over 4 independent accumulators
#define WMMA_BODY(Ab, Bb)                                                            \
    acc0 = __builtin_amdgcn_wmma_f32_16x16x4_f32(false, Ab[0], false, Bb[0],         \
                                                 (short)0, acc0, false, false);      \
    acc1 = __builtin_amdgcn_wmma_f32_16x16x4_f32(false, Ab[1], false, Bb[1],         \
                                                 (short)0, acc1, false, false);      \
    acc2 = __builtin_amdgcn_wmma_f32_16x16x4_f32(false, Ab[2], false, Bb[2],         \
                                                 (short)0, acc2, false, false);      \
    acc3 = __builtin_amdgcn_wmma_f32_16x16x4_f32(false, Ab[3], false, Bb[3],         \
                                                 (short)0, acc3, false, false);      \
    acc0 = __builtin_amdgcn_wmma_f32_16x16x4_f32(false, Ab[4], false, Bb[4],         \
                                                 (short)0, acc0, false, false);      \
    acc1 = __builtin_amdgcn_wmma_f32_16x16x4_f32(false, Ab[5], false, Bb[5],         \
                                                 (short)0, acc1, false, false);      \
    acc2 = __builtin_amdgcn_wmma_f32_16x16x4_f32(false, Ab[6], false, Bb[6],         \
                                                 (short)0, acc2, false, false);      \
    acc3 = __builtin_amdgcn_wmma_f32_16x16x4_f32(false, Ab[7], false, Bb[7],         \
                                                 (short)0, acc3, false, false);

#define LOAD_BODY(Ab, Bb, off)                                                       \
    _Pragma("unroll")                                                                \
    for (int q = 0; q < 8; ++q) {                                                    \
        Ab[q] = *(const v2f*)(aptr  + (off) + 4 * q);                                \
        Bb[q] = *(const v2f*)(bbase + (off) + 4 * q);                                \
    }

// ---------------- persistent RNN kernel ----------------
__global__ __launch_bounds__(THREADS)
void CXBPU_rnn_kernel(const float* __restrict__ x,      // [B,T,I]
                      const float* __restrict__ W_rec,  // [N,N]
                      const float* __restrict__ W_in,   // [NSENS,I]
                      const float* __restrict__ b_in,   // [NSENS]
                      const float* __restrict__ W_out,  // [O,NOUT]
                      const float* __restrict__ b_out,  // [O]
                      const int* __restrict__ out_idx,  // [NOUT] (int32, prebuilt)
                      const int* __restrict__ inv_sens, // [N] col -> sensory slot
                      float* __restrict__ hbuf0,        // [B,N] ping
                      float* __restrict__ hbuf1,        // [B,N] pong
                      unsigned* __restrict__ cnt,       // global barrier counter
                      float* __restrict__ out)          // [B,T,O]
{
    extern __shared__ float Ws[];   // [COLS_PER_BLK][LDS_STRIDE]

    const int tid    = threadIdx.x;
    const int wave   = tid >> 5;
    const int lane   = tid & 31;
    const int lane15 = lane & 15;
    const int hi     = lane >> 4;          // half-wave select (K offset 0 or 2)
    const int j0     = blockIdx.x * COLS_PER_BLK;

    // ---- stage this block's 32-column slice of W_rec into LDS (once) ----
    for (int r = 0; r < COLS_PER_BLK; ++r) {
        const float* src = W_rec + (size_t)(j0 + r) * N_HID;
        float*       dst = Ws + r * LDS_STRIDE;
        for (int c = tid * 4; c < N_HID; c += THREADS * 4) {
            *(v4f*)(dst + c) = *(const v4f*)(src + c);
        }
    }
    __syncthreads();

    // wave -> (M tile of batch, N tile of this block's columns)
    const int mtile = wave & 3;            // 4 tiles of 16 batch rows
    const int ntile = wave >> 2;           // 2 tiles of 16 columns
    const int rowA  = mtile * 16 + lane15; // batch row this lane loads A for
    const int colD  = j0 + ntile * 16 + lane15;

    // B fragment base in LDS: B[k][n] = W_rec[colD][k] ; lane reads k = kk+2*hi, kk+2*hi+1
    const float* bbase = Ws + (ntile * 16 + lane15) * LDS_STRIDE + 2 * hi;

    // Sensory injection setup for this lane's output column
    const int   sidx = inv_sens[colD];
    float w0 = 0.f, w1 = 0.f, w2 = 0.f, w3 = 0.f, bi = 0.f;
    if (sidx >= 0) {
        const float* wr = W_in + sidx * IN_DIM;
        w0 = wr[0]; w1 = wr[1]; w2 = wr[2]; w3 = wr[3];
        bi = b_in[sidx];
    }

    const float* hc = hbuf0;  // read
    float*       hn = hbuf1;  // write
    unsigned target = 0;

    for (int t = 0; t < TSTEPS; ++t) {
        for (int k = 0; k < KMICRO; ++k) {
            // ---- C = relu( H[16 rows] x Wslice[16 cols] ), f32 WMMA,
            //      2-body software pipeline: buffers refilled 2 bodies ahead ----
            v8f acc0 = {0,0,0,0,0,0,0,0};
            v8f acc1 = {0,0,0,0,0,0,0,0};
            v8f acc2 = {0,0,0,0,0,0,0,0};
            v8f acc3 = {0,0,0,0,0,0,0,0};
            const float* aptr = hc + (size_t)rowA * N_HID + 2 * hi;

            v2f A0[8], B0[8], A1[8], B1[8];
            LOAD_BODY(A0, B0, 0);            // body 0
            LOAD_BODY(A1, B1, KBODY);        // body 1

            #pragma unroll 1
            for (int i = 0; i < NBODY - 2; i += 2) {
                WMMA_BODY(A0, B0);
                LOAD_BODY(A0, B0, (i + 2) * KBODY);   // refill for body i+2
                WMMA_BODY(A1, B1);
                LOAD_BODY(A1, B1, (i + 3) * KBODY);   // refill for body i+3
            }
            WMMA_BODY(A0, B0);               // body NBODY-2
            WMMA_BODY(A1, B1);               // body NBODY-1

            v8f acc = (acc0 + acc1) + (acc2 + acc3);

            // ---- sensory injection at first microstep of each timestep ----
            if (k == 0 && sidx >= 0) {
                #pragma unroll
                for (int v = 0; v < 8; ++v) {
                    int m = mtile * 16 + v + 8 * hi;   // batch row of acc[v]
                    const float* xr = x + ((size_t)m * TSTEPS + t) * IN_DIM;
                    acc[v] += bi + xr[0]*w0 + xr[1]*w1 + xr[2]*w2 + xr[3]*w3;
                }
            }

            // ---- relu + store slice of next h ----
            #pragma unroll
            for (int v = 0; v < 8; ++v) {
                int m = mtile * 16 + v + 8 * hi;
                float val = acc[v];
                val = val < 0.0f ? 0.0f : val;
                hn[(size_t)m * N_HID + colD] = val;
            }

            // ---- device-wide barrier (monotonic counter) ----
            __threadfence();
            __syncthreads();
            ++target;
            if (tid == 0) {
                __hip_atomic_fetch_add(cnt, 1u, __ATOMIC_RELEASE,
                                       __HIP_MEMORY_SCOPE_AGENT);
                unsigned want = target * (unsigned)gridDim.x;
                while (__hip_atomic_load(cnt, __ATOMIC_ACQUIRE,
                                         __HIP_MEMORY_SCOPE_AGENT) < want) {
                    __builtin_amdgcn_s_sleep(1);
                }
            }
            __syncthreads();

            // swap ping-pong
            const float* tmp = hn;
            hn = (float*)hc;
            hc = tmp;
        }

        // ---- readout (tiny): block 0 only, 128 (b,o) pairs ----
        if (blockIdx.x == 0 && tid < BATCH * OUT_DIM) {
            int b = tid >> 1;
            int o = tid & 1;
            float s = b_out[o];
            const float* hrow = hc + (size_t)b * N_HID;
            #pragma unroll 4
            for (int i = 0; i < NOUT; ++i) {
                s += hrow[out_idx[i]] * W_out[o * NOUT + i];
            }
            out[((size_t)b * TSTEPS + t) * OUT_DIM + o] = s;
        }
    }
}

// ---------------- launcher ----------------
extern "C" void kernel_launch(void* const* d_in, const int* in_sizes, int n_in,
                              void* d_out, int out_size, void* d_ws, size_t ws_size,
                              hipStream_t stream) {
    (void)in_sizes; (void)n_in; (void)out_size; (void)ws_size;

    const float*     xin   = (const float*)d_in[0];      // [B,T,I]
    const float*     W_rec = (const float*)d_in[1];      // [N,N]
    const float*     W_in  = (const float*)d_in[2];      // [NSENS,I]
    const float*     b_in  = (const float*)d_in[3];      // [NSENS]
    const float*     W_out = (const float*)d_in[4];      // [O,NOUT]
    const float*     b_out = (const float*)d_in[5];      // [O]
    const long long* sens  = (const long long*)d_in[6];  // [NSENS]
    const long long* oidx  = (const long long*)d_in[7];  // [NOUT]
    // d_in[8] = K (==4), hardcoded

    char* ws = (char*)d_ws;
    const size_t HBYTES = (size_t)BATCH * N_HID * sizeof(float);  // 512 KB
    float*    h0     = (float*)ws;
    float*    h1     = (float*)(ws + HBYTES);
    int*      inv    = (int*)(ws + 2 * HBYTES);
    int*      oidx32 = (int*)(ws + 2 * HBYTES + N_HID * sizeof(int));
    unsigned* cnt    = (unsigned*)(ws + 2 * HBYTES + N_HID * sizeof(int)
                                   + NOUT * sizeof(int));

    hipLaunchKernelGGL(CXBPU_init_kernel,
                       dim3((BATCH * N_HID + 255) / 256), dim3(256), 0, stream,
                       h0, cnt, inv, oidx32, oidx);
    hipLaunchKernelGGL(CXBPU_scatter_kernel, dim3(1), dim3(256), 0, stream,
                       sens, inv);

    const size_t ldsBytes = (size_t)COLS_PER_BLK * LDS_STRIDE * sizeof(float); // 262656
    (void)hipFuncSetAttribute((const void*)CXBPU_rnn_kernel,
                              hipFuncAttributeMaxDynamicSharedMemorySize,
                              (int)ldsBytes);

    hipLaunchKernelGGL(CXBPU_rnn_kernel, dim3(NBLK), dim3(THREADS), ldsBytes, stream,
                       xin, W_rec, W_in, b_in, W_out, b_out, oidx32, inv,
                       h0, h1, cnt, (float*)d_out);
}